// Order2_80719615361478
// MI455X (gfx1250) — compile-verified
//
#include <hip/hip_runtime.h>

typedef __attribute__((ext_vector_type(2))) float v2f;
typedef __attribute__((ext_vector_type(4))) float v4f;
typedef __attribute__((ext_vector_type(8))) float v8f;

#define F_DIM 256
#define K_DIM 64
#define OUT_PER_B 32896  // 256*257/2

// Per-batch GEMM: out[b] = A * S, A[f,k] = x[b,f]*W[f,k] (256x64),
// S[k, 64q+r] = A[4k+q, r] (memory-reinterpretation of A).
// Lower-triangular 16x16 tiles only; packed tril store.
__global__ __launch_bounds__(256) void order2_tril_wmma(
    const float* __restrict__ x, const float* __restrict__ W,
    float* __restrict__ out)
{
    // A in LDS, 256 rows x 64 cols f32 = 64 KB, chunk-rotation swizzle:
    // element (row, col) lives at row*64 + (((col>>2)+row)&15)*4 + (col&3).
    // float4 builds aligned, v2f A-frag loads aligned, compute-phase LDS
    // accesses are 64-bank conflict-free.
    __shared__ float As[F_DIM * K_DIM];

    const int b   = blockIdx.x;
    const int tid = threadIdx.x;

    // ---- Phase 1: build A[f,k] = x[b,f] * W[f,k] into swizzled LDS ----
    {
        const int f = tid;                       // 256 threads == 256 rows
        const float xv = x[(size_t)b * F_DIM + f];
        const v4f* wrow = (const v4f*)(W + (size_t)f * K_DIM);
#pragma unroll
        for (int c = 0; c < 16; ++c) {           // 16 float4 chunks per row
            v4f av = wrow[c] * xv;
            *(v4f*)&As[f * 64 + (((c + f) & 15) << 2)] = av;
        }
    }
    __syncthreads();

    // ---- Phase 2: WMMA f32 16x16x4, K=64 per tile ----
    const int wave = tid >> 5;   // 8 waves
    const int lane = tid & 31;
    const int hi   = lane >> 4;  // lane group (0: lanes 0-15, 1: lanes 16-31)
    const int lm   = lane & 15;

    float* outb = out + (size_t)b * OUT_PER_B;

#pragma unroll
    for (int rsel = 0; rsel < 2; ++rsel) {
        // wave w owns tile-rows w and 15-w: (w+1)+(16-w) = 17 tiles/wave
        const int I    = rsel ? (15 - wave) : wave;
        const int arow = 16 * I + lm;

        // Cache all 16 A-fragments for this tile-row in registers.
        // A-matrix 16x4 layout: lanes 0-15: v0=K0,v1=K1; lanes 16-31: v0=K2,v1=K3
        v2f a[16];
#pragma unroll
        for (int t = 0; t < 16; ++t)
            a[t] = *(const v2f*)&As[arow * 64 + (((t + arow) & 15) << 2) + 2 * hi];

        for (int J = 0; J <= I; ++J) {
            const int q     = J >> 2;                       // 64-col block
            const int cbase = ((J & 3) << 2) + (lm >> 2);   // chunk idx of col r0+lm
            const int w     = lm & 3;

            // B rows 4t+k live at A rows 16t + 4k + q.  Since 16t == 0 mod 16,
            // the swizzle rotation is t-invariant: hoist both stream bases and
            // step them by t*16 rows (constant 4096B immediate offsets).
            const int r0     = q + 8 * hi;       // k=0 (lanes0-15) / k=2 (lanes16-31)
            const int r1     = r0 + 4;           // k=1 / k=3
            const float* bp0 = &As[r0 * 64 + (((cbase + r0) & 15) << 2) + w];
            const float* bp1 = &As[r1 * 64 + (((cbase + r1) & 15) << 2) + w];

            v8f acc = {};
#pragma unroll
            for (int t = 0; t < 16; ++t) {
                v2f bf;
                bf.x = bp0[t * 1024];            // +t*16 rows * 64 dwords
                bf.y = bp1[t * 1024];
                acc = __builtin_amdgcn_wmma_f32_16x16x4_f32(
                    false, a[t], false, bf, (short)0, acc, false, false);
            }

            // C/D layout: VGPR v -> M = v + 8*hi, N = lm. Packed tril store.
            const int g = 16 * J + lm;
#pragma unroll
            for (int v = 0; v < 8; ++v) {
                const int f = 16 * I + v + 8 * hi;
                if (J < I || g <= f) {
                    __builtin_nontemporal_store(acc[v],
                        &outb[((f * (f + 1)) >> 1) + g]);
                }
            }
        }
    }
}

extern "C" void kernel_launch(void* const* d_in, const int* in_sizes, int n_in,
                              void* d_out, int out_size, void* d_ws, size_t ws_size,
                              hipStream_t stream) {
    const float* x = (const float*)d_in[0];   // [B, 256] f32
    const float* W = (const float*)d_in[1];   // [256, 64] f32
    float* out = (float*)d_out;               // [B, 32896] f32
    const int B = in_sizes[0] / F_DIM;        // 4096
    order2_tril_wmma<<<B, 256, 0, stream>>>(x, W, out);
}